// QuanvolutionHybrid_65481071403298
// MI455X (gfx1250) — compile-verified
//
#include <hip/hip_runtime.h>
#include <hip/hip_bf16.h>

typedef __attribute__((ext_vector_type(2))) float v2f;
typedef __attribute__((ext_vector_type(8))) float v8f;

__device__ __forceinline__ v8f wmma_f32_4(v2f a, v2f b, v8f c) {
    // D = A(16x4 f32) * B(4x16 f32) + C(16x16 f32)
    return __builtin_amdgcn_wmma_f32_16x16x4_f32(
        /*neg_a=*/false, a, /*neg_b=*/false, b,
        /*c_mod=*/(short)0, c, /*reuse_a=*/false, /*reuse_b=*/false);
}

// ---------------------------------------------------------------------------
// Kernel A: build the fixed 16x16 circuit unitary U (Ur, Ui) from rand_params.
// 16 lanes, lane t evolves basis state |t>.  wire w <-> bit (3-w).
// ---------------------------------------------------------------------------
__device__ __forceinline__ void gate1(float* re, float* im, int bit,
                                      float u00r, float u00i, float u01r, float u01i,
                                      float u10r, float u10i, float u11r, float u11i) {
    const int mask = 1 << bit;
#pragma unroll
    for (int i = 0; i < 16; ++i) {
        if (i & mask) continue;
        const int j = i | mask;
        float ar = re[i], ai = im[i], br = re[j], bi = im[j];
        re[i] = u00r * ar - u00i * ai + u01r * br - u01i * bi;
        im[i] = u00r * ai + u00i * ar + u01r * bi + u01i * br;
        re[j] = u10r * ar - u10i * ai + u11r * br - u11i * bi;
        im[j] = u10r * ai + u10i * ar + u11r * bi + u11i * br;
    }
}

__device__ __forceinline__ void cnotg(float* re, float* im, int cbit, int tbit) {
    const int cm = 1 << cbit, tm = 1 << tbit;
#pragma unroll
    for (int i = 0; i < 16; ++i) {
        if ((i & cm) && !(i & tm)) {
            const int j = i | tm;
            float tr = re[i]; re[i] = re[j]; re[j] = tr;
            float ti = im[i]; im[i] = im[j]; im[j] = ti;
        }
    }
}

__global__ void precompute_unitary(const float* __restrict__ rp,
                                   float* __restrict__ Mr, float* __restrict__ Mi) {
    const int t = threadIdx.x;
    if (t >= 16) return;
    float re[16], im[16];
#pragma unroll
    for (int i = 0; i < 16; ++i) { re[i] = 0.f; im[i] = 0.f; }
    re[t] = 1.f;

    float c, s;
    // 1. Ry(rp0) on wire0 (bit 3)
    s = sinf(rp[0] * 0.5f); c = cosf(rp[0] * 0.5f);
    gate1(re, im, 3, c, 0.f, -s, 0.f, s, 0.f, c, 0.f);
    // 2. Rx(rp1) on wire1 (bit 2)
    s = sinf(rp[1] * 0.5f); c = cosf(rp[1] * 0.5f);
    gate1(re, im, 2, c, 0.f, 0.f, -s, 0.f, -s, c, 0.f);
    // 3. CNOT(0,1): ctrl bit3, tgt bit2
    cnotg(re, im, 3, 2);
    // 4. Rz(rp2) on wire2 (bit 1): diag(e^{-it/2}, e^{+it/2})
    s = sinf(rp[2] * 0.5f); c = cosf(rp[2] * 0.5f);
    gate1(re, im, 1, c, -s, 0.f, 0.f, 0.f, 0.f, c, s);
    // 5. Ry(rp3) on wire3 (bit 0)
    s = sinf(rp[3] * 0.5f); c = cosf(rp[3] * 0.5f);
    gate1(re, im, 0, c, 0.f, -s, 0.f, s, 0.f, c, 0.f);
    // 6. CNOT(2,3): ctrl bit1, tgt bit0
    cnotg(re, im, 1, 0);
    // 7. Rx(rp4) on wire0 (bit 3)
    s = sinf(rp[4] * 0.5f); c = cosf(rp[4] * 0.5f);
    gate1(re, im, 3, c, 0.f, 0.f, -s, 0.f, -s, c, 0.f);
    // 8. CNOT(1,2): ctrl bit2, tgt bit1
    cnotg(re, im, 2, 1);

#pragma unroll
    for (int i = 0; i < 16; ++i) {
        Mr[i * 16 + t] = re[i];
        Mi[i * 16 + t] = im[i];
    }
}

// ---------------------------------------------------------------------------
// Kernel B: per wave, simulate 16 patches via WMMA f32 16x16x4.
//   B operand = Psi (16 amplitudes x 16 patches), A = Ur / Ui tiles.
//   Cr = Ur*Psi, Ci = Ui*Psi;  p = Cr^2+Ci^2;  e_w = sign-weighted sums of p.
// ---------------------------------------------------------------------------
__global__ __launch_bounds__(256) void quanv_kernel(const float* __restrict__ x,
                                                    const float* __restrict__ Mr,
                                                    const float* __restrict__ Mi,
                                                    float* __restrict__ feats) {
    const int tid  = blockIdx.x * 256 + (int)threadIdx.x;
    const int wave = tid >> 5;
    const int lane = threadIdx.x & 31;
    const int half = lane >> 4;     // 0: lanes 0-15, 1: lanes 16-31
    const int l    = lane & 15;
    const int n    = wave * 16 + l; // global patch id (grid covers exactly 802816)

    const int b  = n / 196;
    const int p  = n - b * 196;
    const int pr = p / 14;
    const int pc = p - pr * 14;
    const float* xb = x + b * 784 + pr * 56 + pc * 2;

    float s0, c0, s1, c1, s2, c2, s3, c3;
    __sincosf(xb[0]  * 0.5f, &s0, &c0);   // wire0 = (r,c)
    __sincosf(xb[1]  * 0.5f, &s1, &c1);   // wire1 = (r,c+1)
    __sincosf(xb[28] * 0.5f, &s2, &c2);   // wire2 = (r+1,c)
    __sincosf(xb[29] * 0.5f, &s3, &c3);   // wire3 = (r+1,c+1)

    // psi[j] = w0[j>>3]*w1[(j>>2)&1]*w2[(j>>1)&1]*w3[j&1]
    // B chunk k, vgpr0 holds Psi row 4k+2*half, vgpr1 holds row 4k+2*half+1.
    const float w2v = half ? s2 : c2;
    const float q0 = c0 * c1 * w2v;   // k=0 (bits3:2 = 00)
    const float q1 = c0 * s1 * w2v;   // k=1
    const float q2 = s0 * c1 * w2v;   // k=2
    const float q3 = s0 * s1 * w2v;   // k=3
    v2f B0 = {q0 * c3, q0 * s3};
    v2f B1 = {q1 * c3, q1 * s3};
    v2f B2 = {q2 * c3, q2 * s3};
    v2f B3 = {q3 * c3, q3 * s3};

    // A chunk k: lane row = l, cols 4k+2*half, 4k+2*half+1 (row-major 16x16)
    const int aoff = l * 16 + 2 * half;
    v2f Ar0 = *(const v2f*)(Mr + aoff);
    v2f Ar1 = *(const v2f*)(Mr + aoff + 4);
    v2f Ar2 = *(const v2f*)(Mr + aoff + 8);
    v2f Ar3 = *(const v2f*)(Mr + aoff + 12);
    v2f Ai0 = *(const v2f*)(Mi + aoff);
    v2f Ai1 = *(const v2f*)(Mi + aoff + 4);
    v2f Ai2 = *(const v2f*)(Mi + aoff + 8);
    v2f Ai3 = *(const v2f*)(Mi + aoff + 12);

    v8f cr = {};
    cr = wmma_f32_4(Ar0, B0, cr);
    cr = wmma_f32_4(Ar1, B1, cr);
    cr = wmma_f32_4(Ar2, B2, cr);
    cr = wmma_f32_4(Ar3, B3, cr);
    v8f ci = {};
    ci = wmma_f32_4(Ai0, B0, ci);
    ci = wmma_f32_4(Ai1, B1, ci);
    ci = wmma_f32_4(Ai2, B2, ci);
    ci = wmma_f32_4(Ai3, B3, ci);

    // D layout: vgpr r, this lane -> amplitude i = r + 8*half, patch = l
    float e0 = 0.f, e1 = 0.f, e2 = 0.f, e3 = 0.f;
#pragma unroll
    for (int r = 0; r < 8; ++r) {
        const float pv = cr[r] * cr[r] + ci[r] * ci[r];
        e0 += pv;                                  // bit3 sign folded in below
        e1 += ((r >> 2) & 1) ? -pv : pv;           // wire1 <-> bit2
        e2 += ((r >> 1) & 1) ? -pv : pv;           // wire2 <-> bit1
        e3 += (r & 1) ? -pv : pv;                  // wire3 <-> bit0
    }
    if (half) e0 = -e0;                            // wire0 <-> bit3 == half

    e0 += __shfl_xor(e0, 16, 32);
    e1 += __shfl_xor(e1, 16, 32);
    e2 += __shfl_xor(e2, 16, 32);
    e3 += __shfl_xor(e3, 16, 32);

    if (!half) {
        float4 out = make_float4(e0, e1, e2, e3);
        *(float4*)(feats + (size_t)n * 4) = out;   // feats[b][p*4 + w]
    }
}

// ---------------------------------------------------------------------------
// Kernel C: logits = feats(4096x784) @ W^T(784x10) + b, via WMMA 16x16x4.
// One wave per 16 batch rows; N=16 cols, lanes with col>=10 use zeroed B.
// ---------------------------------------------------------------------------
__global__ __launch_bounds__(256) void logits_kernel(const float* __restrict__ feats,
                                                     const float* __restrict__ W,
                                                     const float* __restrict__ bias,
                                                     float* __restrict__ logits) {
    const int tid  = blockIdx.x * 256 + (int)threadIdx.x;
    const int wave = tid >> 5;
    const int lane = threadIdx.x & 31;
    const int half = lane >> 4;
    const int l    = lane & 15;
    const int row  = wave * 16 + l;

    const bool  bvalid = (l < 10);
    const float bscale = bvalid ? 1.f : 0.f;
    const float* arow = feats + (size_t)row * 784 + 2 * half;
    const float* brow = W + (size_t)(bvalid ? l : 0) * 784 + 2 * half;

    v8f acc = {};
    for (int k = 0; k < 784; k += 4) {
        v2f a  = *(const v2f*)(arow + k);
        v2f bv = *(const v2f*)(brow + k);
        bv.x *= bscale;
        bv.y *= bscale;
        acc = wmma_f32_4(a, bv, acc);
    }

    if (bvalid) {
        const float bb = bias[l];
#pragma unroll
        for (int r = 0; r < 8; ++r) {
            const int m = wave * 16 + r + 8 * half;
            logits[m * 10 + l] = acc[r] + bb;
        }
    }
}

// ---------------------------------------------------------------------------
// Kernel D: row-wise log_softmax over 10 classes.
// ---------------------------------------------------------------------------
__global__ void logsoftmax_kernel(const float* __restrict__ logits,
                                  float* __restrict__ out) {
    const int row = blockIdx.x * blockDim.x + threadIdx.x;
    if (row >= 4096) return;
    float v[10];
    float m = -3.0e38f;
#pragma unroll
    for (int i = 0; i < 10; ++i) {
        v[i] = logits[row * 10 + i];
        m = fmaxf(m, v[i]);
    }
    float s = 0.f;
#pragma unroll
    for (int i = 0; i < 10; ++i) s += expf(v[i] - m);
    const float lse = m + logf(s);
#pragma unroll
    for (int i = 0; i < 10; ++i) out[row * 10 + i] = v[i] - lse;
}

// ---------------------------------------------------------------------------
extern "C" void kernel_launch(void* const* d_in, const int* in_sizes, int n_in,
                              void* d_out, int out_size, void* d_ws, size_t ws_size,
                              hipStream_t stream) {
    const float* x    = (const float*)d_in[0];   // (4096,28,28)
    const float* rp   = (const float*)d_in[1];   // (5,)
    const float* W    = (const float*)d_in[2];   // (10,784)
    const float* bias = (const float*)d_in[3];   // (10,)
    float* out = (float*)d_out;                  // (4096,10)

    float* ws     = (float*)d_ws;
    float* Mr     = ws;                          // 256 floats
    float* Mi     = ws + 256;                    // 256 floats
    float* feats  = ws + 512;                    // 802816*4 = 3,211,264 floats
    float* logits = ws + 512 + 3211264;          // 4096*10 floats

    precompute_unitary<<<1, 32, 0, stream>>>(rp, Mr, Mi);
    // 802816 patches / 16 per wave = 50176 waves; 8 waves/block -> 6272 blocks
    quanv_kernel<<<6272, 256, 0, stream>>>(x, Mr, Mi, feats);
    // 4096 rows / 16 per wave = 256 waves; 8 waves/block -> 32 blocks
    logits_kernel<<<32, 256, 0, stream>>>(feats, W, bias, logits);
    logsoftmax_kernel<<<16, 256, 0, stream>>>(logits, out);
}